// GCN_75179107549523
// MI455X (gfx1250) — compile-verified
//
#include <hip/hip_runtime.h>
#include <cstdint>

typedef float v2f __attribute__((ext_vector_type(2)));
typedef float v8f __attribute__((ext_vector_type(8)));

#define GCN_EPT 16  // edges per scatter wave (dst sorted, avg run length ~16)

// ---- optional gfx1250 async-to-LDS path (probe via __has_builtin) ----------
#if defined(__HIP_DEVICE_COMPILE__) && defined(__gfx1250__) && \
    __has_builtin(__builtin_amdgcn_global_load_async_to_lds_b128) && \
    __has_builtin(__builtin_amdgcn_s_wait_asynccnt)
#define GCN_ASYNC_LDS 1
// ROCm 7.2 clang-22 signature (from diagnostics): param 1 is a GENERIC pointer
// to int32x4; pass the LDS side as an addrspace(3) int32x4* (implicitly
// convertible if the param is generic or void-typed).
typedef int gcn_v4i __attribute__((vector_size(4 * sizeof(int))));
typedef __attribute__((address_space(3))) gcn_v4i* gcn_lds_v4i_t;
#else
#define GCN_ASYNC_LDS 0
#endif

// ---------------------------------------------------------------------------
// GEMM: C[N x KO] = A[N x 128] * W[128 x KO]  (fp32, V_WMMA_F32_16X16X4_F32)
// One 64-row block per workgroup; A staged in LDS (async DMA on the full-block
// fast path); W read via L2 (64KB, hot across all blocks). Each wave owns one
// 16-wide column tile -> B fragment loaded once per k-step, reused across its
// row tiles.
// ---------------------------------------------------------------------------
template <int KO>
__global__ __launch_bounds__(256) void gcn_gemm_wmma(
    const float* __restrict__ A, const float* __restrict__ W,
    float* __restrict__ C, int N)
{
    constexpr int ROWS       = 64;
    constexpr int numCT      = KO / 16;                  // 8 (KO=128) or 4 (KO=64)
    constexpr int wavesPerCT = 8 / numCT;                // 1 or 2
    constexpr int rtPerWave  = (ROWS / 16) / wavesPerCT; // 4 or 2

    __shared__ float ldsA[ROWS * 128];                   // 32 KB

    const int  row0 = blockIdx.x * ROWS;
    const bool full = (row0 + ROWS) <= N;                // block-uniform

    __builtin_prefetch(W, 0, 3);                         // global_prefetch_b8

    if (full) {
        // fast path: no guards; async DMA straight into LDS when available
        for (int i = threadIdx.x; i < ROWS * 32; i += 256) {
            const int r  = i >> 5;
            const int c4 = (i & 31) << 2;
#if GCN_ASYNC_LDS
            __builtin_amdgcn_global_load_async_to_lds_b128(
                (gcn_v4i*)(A + (size_t)(row0 + r) * 128 + c4),
                (gcn_lds_v4i_t)(uintptr_t)(&ldsA[r * 128 + c4]),
                0, 0);
#else
            *(float4*)(&ldsA[r * 128 + c4]) =
                *(const float4*)(A + (size_t)(row0 + r) * 128 + c4);
#endif
        }
#if GCN_ASYNC_LDS
        __builtin_amdgcn_s_wait_asynccnt(0);
#endif
    } else {
        for (int i = threadIdx.x; i < ROWS * 32; i += 256) {
            const int r  = i >> 5;
            const int c4 = (i & 31) << 2;
            float4 v = make_float4(0.f, 0.f, 0.f, 0.f);
            if (row0 + r < N)
                v = *(const float4*)(A + (size_t)(row0 + r) * 128 + c4);
            *(float4*)(&ldsA[r * 128 + c4]) = v;
        }
    }
    __syncthreads();

    const int wave = threadIdx.x >> 5;
    const int lane = threadIdx.x & 31;
    const int hl   = lane >> 4;    // half-wave: selects K pair (A) / K row pair (B)
    const int l16  = lane & 15;

    const int ct  = wave / wavesPerCT;
    const int rt0 = (wave % wavesPerCT) * rtPerWave;

    v8f zero = {};
    v8f acc[rtPerWave];
#pragma unroll
    for (int t = 0; t < rtPerWave; ++t) acc[t] = zero;

#pragma unroll 4
    for (int k = 0; k < 128; k += 4) {
        const int kk = k + 2 * hl;
        // B 4x16: VGPR0 = row kk, VGPR1 = row kk+1, N = l16 (striped across lanes)
        v2f b;
        b.x = W[(size_t)kk * KO + ct * 16 + l16];
        b.y = W[(size_t)(kk + 1) * KO + ct * 16 + l16];
#pragma unroll
        for (int t = 0; t < rtPerWave; ++t) {
            // A 16x4: lane row = l16, cols kk,kk+1 (contiguous -> one b64 LDS load)
            const v2f a = *(const v2f*)(&ldsA[((rt0 + t) * 16 + l16) * 128 + kk]);
            acc[t] = __builtin_amdgcn_wmma_f32_16x16x4_f32(
                false, a, false, b, (short)0, acc[t], false, false);
        }
    }

    // C/D 16x16 f32 layout: VGPR j -> row j (lanes 0-15) / row j+8 (lanes 16-31)
    if (full) {
#pragma unroll
        for (int t = 0; t < rtPerWave; ++t) {
            const int rt = rt0 + t;
#pragma unroll
            for (int j = 0; j < 8; ++j) {
                const int row = row0 + rt * 16 + j + 8 * hl;
                C[(size_t)row * KO + ct * 16 + l16] = acc[t][j];
            }
        }
    } else {
#pragma unroll
        for (int t = 0; t < rtPerWave; ++t) {
            const int rt = rt0 + t;
#pragma unroll
            for (int j = 0; j < 8; ++j) {
                const int row = row0 + rt * 16 + j + 8 * hl;
                if (row < N)
                    C[(size_t)row * KO + ct * 16 + l16] = acc[t][j];
            }
        }
    }
}

// ---------------------------------------------------------------------------
// agg[n, c] = bias[c]   (bias added before scatter-accumulation)
// ---------------------------------------------------------------------------
__global__ void gcn_init_bias(float* __restrict__ out, const float* __restrict__ b,
                              long total, int cmask)
{
    const long i = (long)blockIdx.x * blockDim.x + threadIdx.x;
    if (i < total) out[i] = b[i & cmask];
}

// ---------------------------------------------------------------------------
// Segmented scatter-add over sorted dst, one WAVE per edge chunk:
//  - all 32 lanes cooperatively hold one full C-wide message row
//    (C=128 -> float4/lane = coalesced 512B load per edge)
//  - run accumulation in registers; dst-change branch is wave-uniform
//  - flush = coalesced burst of global_atomic_add_f32 at run boundaries only
// ---------------------------------------------------------------------------
template <int C>
__global__ __launch_bounds__(256) void gcn_scatter_add(
    const float* __restrict__ m, float* __restrict__ agg,
    const int* __restrict__ src, const int* __restrict__ dst, int E)
{
    constexpr int VW = C / 32;   // floats per lane: 4 (C=128) or 2 (C=64)
    const int  wid  = (blockIdx.x * blockDim.x + threadIdx.x) >> 5; // global wave
    const int  lane = threadIdx.x & 31;
    const long e0   = (long)wid * GCN_EPT;
    if (e0 >= E) return;
    const long e1  = (e0 + GCN_EPT < E) ? (e0 + GCN_EPT) : (long)E;
    const int  col = lane * VW;

    float acc[VW];
#pragma unroll
    for (int i = 0; i < VW; ++i) acc[i] = 0.f;

    int cur = dst[e0];                          // wave-uniform
    for (long e = e0; e < e1; ++e) {
        const int d = dst[e];                   // wave-uniform scalar load
        if (d != cur) {                         // wave-uniform branch
            float* p = agg + (size_t)cur * C + col;
#pragma unroll
            for (int i = 0; i < VW; ++i) { unsafeAtomicAdd(p + i, acc[i]); acc[i] = 0.f; }
            cur = d;
        }
        const float* row = m + (size_t)src[e] * C + col;
        if constexpr (VW == 4) {
            const float4 v = *(const float4*)row;
            acc[0] += v.x; acc[1] += v.y; acc[2] += v.z; acc[3] += v.w;
        } else {
            const float2 v = *(const float2*)row;
            acc[0] += v.x; acc[1] += v.y;
        }
    }
    float* p = agg + (size_t)cur * C + col;
#pragma unroll
    for (int i = 0; i < VW; ++i) unsafeAtomicAdd(p + i, acc[i]);
}

__global__ void gcn_zero(float* __restrict__ p, int n)
{
    const int i = blockIdx.x * blockDim.x + threadIdx.x;
    if (i < n) p[i] = 0.f;
}

// ---------------------------------------------------------------------------
// Per-channel sum / sum-of-squares over N rows (C == 128, one channel/thread)
// ---------------------------------------------------------------------------
__global__ __launch_bounds__(128) void gcn_bn_stats(
    const float* __restrict__ h, float* __restrict__ stats, int N)
{
    const int c = threadIdx.x;
    float s = 0.f, sq = 0.f;
    for (int r = blockIdx.x; r < N; r += gridDim.x) {
        const float v = h[(size_t)r * 128 + c];
        s += v; sq += v * v;
    }
    unsafeAtomicAdd(&stats[c], s);
    unsafeAtomicAdd(&stats[128 + c], sq);
}

// ---------------------------------------------------------------------------
// out = relu( (agg - mu) * rsqrt(var+eps) * gamma + beta + residual )
// ---------------------------------------------------------------------------
__global__ void gcn_bn_apply(
    const float* __restrict__ agg, const float* __restrict__ res,
    const float* __restrict__ stats, const float* __restrict__ gamma,
    const float* __restrict__ beta, float* __restrict__ out, int N)
{
    const long i = (long)blockIdx.x * blockDim.x + threadIdx.x;
    if (i >= (long)N * 128) return;
    const int   c    = (int)(i & 127);
    const float invN = 1.0f / (float)N;
    const float mu   = stats[c] * invN;
    const float var  = stats[128 + c] * invN - mu * mu;
    const float scl  = rsqrtf(var + 1e-5f) * gamma[c];
    const float v    = (agg[i] - mu) * scl + beta[c] + res[i];
    out[i] = v > 0.f ? v : 0.f;
}

// ---------------------------------------------------------------------------
extern "C" void kernel_launch(void* const* d_in, const int* in_sizes, int n_in,
                              void* d_out, int out_size, void* d_ws, size_t ws_size,
                              hipStream_t stream)
{
    const float* x   = (const float*)d_in[0];
    const int*   src = (const int*)  d_in[1];
    const int*   dst = (const int*)  d_in[2];
    const float* W0  = (const float*)d_in[3];
    const float* b0  = (const float*)d_in[4];
    const float* W1  = (const float*)d_in[5];
    const float* b1  = (const float*)d_in[6];
    const float* W2  = (const float*)d_in[7];
    const float* b2  = (const float*)d_in[8];
    const float* g0  = (const float*)d_in[9];
    const float* be0 = (const float*)d_in[10];
    const float* g1  = (const float*)d_in[11];
    const float* be1 = (const float*)d_in[12];

    const int N = in_sizes[0] / 128;
    const int E = in_sizes[1];

    float* m     = (float*)d_ws;                 // N x 128
    float* agg   = m   + (size_t)N * 128;        // N x 128
    float* h     = agg + (size_t)N * 128;        // N x 128 (activations, reused)
    float* stats = h   + (size_t)N * 128;        // 256 floats
    float* out   = (float*)d_out;                // N x 64

    const int  gemmBlocks = (N + 63) / 64;
    const int  waveChunks = (E + GCN_EPT - 1) / GCN_EPT;  // one wave per chunk
    const int  scatBlocks = (waveChunks + 7) / 8;         // 8 waves / block
    const long tot128     = (long)N * 128;
    const int  ewBlocks   = (int)((tot128 + 255) / 256);
    const long tot64      = (long)N * 64;
    const int  ewBlocks64 = (int)((tot64 + 255) / 256);

    // ---- layer 0: conv -> BN -> +x -> relu ----
    gcn_gemm_wmma<128><<<gemmBlocks, 256, 0, stream>>>(x, W0, m, N);
    gcn_init_bias<<<ewBlocks, 256, 0, stream>>>(agg, b0, tot128, 127);
    gcn_scatter_add<128><<<scatBlocks, 256, 0, stream>>>(m, agg, src, dst, E);
    gcn_zero<<<1, 256, 0, stream>>>(stats, 256);
    gcn_bn_stats<<<512, 128, 0, stream>>>(agg, stats, N);
    gcn_bn_apply<<<ewBlocks, 256, 0, stream>>>(agg, x, stats, g0, be0, h, N);

    // ---- layer 1: conv -> BN -> +x1 -> relu (residual applied in place) ----
    gcn_gemm_wmma<128><<<gemmBlocks, 256, 0, stream>>>(h, W1, m, N);
    gcn_init_bias<<<ewBlocks, 256, 0, stream>>>(agg, b1, tot128, 127);
    gcn_scatter_add<128><<<scatBlocks, 256, 0, stream>>>(m, agg, src, dst, E);
    gcn_zero<<<1, 256, 0, stream>>>(stats, 256);
    gcn_bn_stats<<<512, 128, 0, stream>>>(agg, stats, N);
    gcn_bn_apply<<<ewBlocks, 256, 0, stream>>>(agg, h, stats, g1, be1, h, N);

    // ---- layer 2: conv only, aggregate straight into d_out ----
    gcn_gemm_wmma<64><<<gemmBlocks, 256, 0, stream>>>(h, W2, m, N);
    gcn_init_bias<<<ewBlocks64, 256, 0, stream>>>(out, b2, tot64, 63);
    gcn_scatter_add<64><<<scatBlocks, 256, 0, stream>>>(m, out, src, dst, E);
}